// SparseGaussianHead_38843684225421
// MI455X (gfx1250) — compile-verified
//
#include <hip/hip_runtime.h>
#include <hip/hip_bf16.h>
#include <math.h>

typedef __attribute__((ext_vector_type(2))) float v2f;
typedef __attribute__((ext_vector_type(4))) float v4f;
typedef __attribute__((ext_vector_type(8))) float v8f;

#define NVOX 400000
#define KOFF 27
#define C_IN 128
#define C_MID 14
#define MSTR 16                            // padded channel stride
#define TILES (NVOX / 16)                  // 25000 (N divisible by 16)
#define WPB 8                              // 256 threads = 8 wave32

static_assert(NVOX % 16 == 0, "tile cover");

// Workspace layout (float offsets)
#define WS_MID    0                                   // (N+2) x 16 : conv1 acc; row N = zeros, row N+1 = trash
#define WS_OUT16  (WS_MID + (NVOX + 2) * MSTR)        // (N+1) x 16 : conv2 acc; row N = trash
#define WS_ZROW   (WS_OUT16 + (NVOX + 1) * MSTR)      // 128 zeros (dummy feat row)
#define WS_BSWZ1  ((WS_ZROW + C_IN + 15) & ~15)       // 27*32*32*2 swizzled padded kernel1 fragments
#define WS_BSWZ2  (WS_BSWZ1 + KOFF * 32 * 32 * 2)     // 27*4*32*2  swizzled padded kernel2 fragments
#define WS_END    (WS_BSWZ2 + KOFF * 4 * 32 * 2)
#define WS_ZERO_FLOATS WS_BSWZ1                       // memset range start..

// ---------------------------------------------------------------------------
// Pre-swizzle kernel1 [27][128][14] -> B fragments [27][32 steps][32 lanes][2]
// lane l: half = l>>4, n = l&15, kk = 4t + 2*half ; pad n>=14 with zero.
// ---------------------------------------------------------------------------
__global__ __launch_bounds__(256)
void sgh_prep_b1(const float* __restrict__ w1, float* __restrict__ bswz1)
{
    const int i = blockIdx.x * 256 + threadIdx.x;     // [0, 27*32*32)
    if (i >= KOFF * 32 * 32) return;
    const int lane = i & 31, t = (i >> 5) & 31, k = i >> 10;
    const int half = lane >> 4, n = lane & 15, kk = (t << 2) + (half << 1);
    const bool nv = (n < C_MID);
    bswz1[2 * i]     = nv ? w1[((size_t)k * C_IN + kk)     * C_MID + n] : 0.f;
    bswz1[2 * i + 1] = nv ? w1[((size_t)k * C_IN + kk + 1) * C_MID + n] : 0.f;
}

// Pre-swizzle kernel2 [27][14][14] -> [27][4][32][2], pad n>=14 and kk>=14.
__global__ __launch_bounds__(256)
void sgh_prep_b2(const float* __restrict__ w2, float* __restrict__ bswz2)
{
    const int i = blockIdx.x * 256 + threadIdx.x;     // [0, 27*4*32)
    if (i >= KOFF * 4 * 32) return;
    const int lane = i & 31, t = (i >> 5) & 3, k = i >> 7;
    const int half = lane >> 4, n = lane & 15, kk = (t << 2) + (half << 1);
    const bool nv = (n < C_MID);
    bswz2[2 * i]     = (nv && kk     < C_MID) ? w2[((size_t)k * C_MID + kk)     * C_MID + n] : 0.f;
    bswz2[2 * i + 1] = (nv && kk + 1 < C_MID) ? w2[((size_t)k * C_MID + kk + 1) * C_MID + n] : 0.f;
}

// ---------------------------------------------------------------------------
// Conv1: one wave32 per (k, 16-voxel tile). 32 x V_WMMA_F32_16X16X4_F32,
// branchless gathers (pointer select to zero row), unconditional scatter-add
// with trash-row redirect for dummy indices.
// ---------------------------------------------------------------------------
__global__ __launch_bounds__(256)
void sgh_conv1_wmma(const float* __restrict__ feat,     // [N][128]
                    const float* __restrict__ bswz1,    // [27][32][32][2]
                    const float* __restrict__ zrow,     // [128] zeros
                    const int*   __restrict__ in_idx,   // [27][N]
                    const int*   __restrict__ out_idx,  // [27][N]
                    float*       __restrict__ mid)      // [(N+2)][16]
{
    const int wave = blockIdx.x * WPB + (threadIdx.x >> 5);
    const int lane = threadIdx.x & 31;
    const int k    = wave / TILES;
    const int tile = wave - k * TILES;
    if (k >= KOFF) return;                              // wave-uniform

    const int row16 = lane & 15;
    const int half  = lane >> 4;
    const int base  = tile << 4;

    const int aidx = in_idx[(size_t)k * NVOX + base + row16];
    const float* arow = (aidx < NVOX) ? feat + (size_t)aidx * C_IN : zrow;  // branchless
    const float* bp   = bswz1 + (((size_t)k * 32) * 32 + lane) * 2;         // +64 floats/step

    v8f c = {0.f, 0.f, 0.f, 0.f, 0.f, 0.f, 0.f, 0.f};
    v2f a = *(const v2f*)(arow + (half << 1));
    v2f b = *(const v2f*)(bp);

    #pragma unroll
    for (int t = 0; t < C_IN / 4; ++t) {
        v2f an = a, bn = b;
        if (t + 1 < C_IN / 4) {                          // prefetch next step
            an = *(const v2f*)(arow + ((t + 1) << 2) + (half << 1));
            bn = *(const v2f*)(bp + (size_t)(t + 1) * 64);
        }
        c = __builtin_amdgcn_wmma_f32_16x16x4_f32(false, a, false, b,
                                                  (short)0, c, false, false);
        a = an; b = bn;
    }

    // Scatter-add all 16 columns (pad cols are exactly 0); dummy -> trash row.
    const int* oo = out_idx + (size_t)k * NVOX + base + (half << 3);
    const int4 o0 = *(const int4*)(oo);
    const int4 o1 = *(const int4*)(oo + 4);
    const int ors[8] = {o0.x, o0.y, o0.z, o0.w, o1.x, o1.y, o1.z, o1.w};
    #pragma unroll
    for (int j = 0; j < 8; ++j) {
        const int eff = (ors[j] < NVOX) ? ors[j] : (NVOX + 1);
        atomicAdd(mid + (size_t)eff * MSTR + row16, c[j]);
    }
}

// ---------------------------------------------------------------------------
// Exact GELU, vectorized x4, unconditional (gelu(0)=0 keeps pad cols zero).
// ---------------------------------------------------------------------------
__global__ __launch_bounds__(256)
void sgh_gelu(float* __restrict__ mid)
{
    const size_t i = ((size_t)blockIdx.x * 256 + threadIdx.x) * 4;
    if (i >= (size_t)NVOX * MSTR) return;
    v4f x = *(const v4f*)(mid + i);
    v4f y;
    #pragma unroll
    for (int j = 0; j < 4; ++j)
        y[j] = 0.5f * x[j] * (1.0f + erff(x[j] * 0.70710678118654752f));
    *(v4f*)(mid + i) = y;
}

// ---------------------------------------------------------------------------
// Conv2: 4 x V_WMMA_F32_16X16X4_F32; dummy gathers read zero row N of mid.
// ---------------------------------------------------------------------------
__global__ __launch_bounds__(256)
void sgh_conv2_wmma(const float* __restrict__ mid,      // [(N+2)][16], gelu'd
                    const float* __restrict__ bswz2,    // [27][4][32][2]
                    const int*   __restrict__ in_idx,
                    const int*   __restrict__ out_idx,
                    float*       __restrict__ out16)    // [(N+1)][16]
{
    const int wave = blockIdx.x * WPB + (threadIdx.x >> 5);
    const int lane = threadIdx.x & 31;
    const int k    = wave / TILES;
    const int tile = wave - k * TILES;
    if (k >= KOFF) return;

    const int row16 = lane & 15;
    const int half  = lane >> 4;
    const int base  = tile << 4;

    const int aidx = in_idx[(size_t)k * NVOX + base + row16];   // idx==N -> zero row
    const float* arow = mid + (size_t)aidx * MSTR;
    const float* bp   = bswz2 + (((size_t)k * 4) * 32 + lane) * 2;

    v8f c = {0.f, 0.f, 0.f, 0.f, 0.f, 0.f, 0.f, 0.f};
    v2f a = *(const v2f*)(arow + (half << 1));
    v2f b = *(const v2f*)(bp);

    #pragma unroll
    for (int t = 0; t < 4; ++t) {
        v2f an = a, bn = b;
        if (t + 1 < 4) {
            an = *(const v2f*)(arow + ((t + 1) << 2) + (half << 1));
            bn = *(const v2f*)(bp + (size_t)(t + 1) * 64);
        }
        c = __builtin_amdgcn_wmma_f32_16x16x4_f32(false, a, false, b,
                                                  (short)0, c, false, false);
        a = an; b = bn;
    }

    const int* oo = out_idx + (size_t)k * NVOX + base + (half << 3);
    const int4 o0 = *(const int4*)(oo);
    const int4 o1 = *(const int4*)(oo + 4);
    const int ors[8] = {o0.x, o0.y, o0.z, o0.w, o1.x, o1.y, o1.z, o1.w};
    #pragma unroll
    for (int j = 0; j < 8; ++j) {
        const int eff = (ors[j] < NVOX) ? ors[j] : NVOX;         // trash row N
        atomicAdd(out16 + (size_t)eff * MSTR + row16, c[j]);
    }
}

// ---------------------------------------------------------------------------
// Repack stride-16 accumulator into the dense [N][14] output.
// ---------------------------------------------------------------------------
__global__ __launch_bounds__(256)
void sgh_repack(const float* __restrict__ out16, float* __restrict__ out)
{
    const int i = blockIdx.x * 256 + threadIdx.x;
    if (i >= NVOX * C_MID) return;
    const int r = i / C_MID;
    const int cidx = i - r * C_MID;
    out[i] = out16[(size_t)r * MSTR + cidx];
}

// ---------------------------------------------------------------------------
extern "C" void kernel_launch(void* const* d_in, const int* in_sizes, int n_in,
                              void* d_out, int out_size, void* d_ws, size_t ws_size,
                              hipStream_t stream)
{
    const float* feat    = (const float*)d_in[0];
    const float* kernel1 = (const float*)d_in[1];
    const float* kernel2 = (const float*)d_in[2];
    const int*   in_idx  = (const int*)d_in[3];
    const int*   out_idx = (const int*)d_in[4];
    float*       out     = (float*)d_out;

    float* ws    = (float*)d_ws;
    float* mid   = ws + WS_MID;
    float* out16 = ws + WS_OUT16;
    float* zrow  = ws + WS_ZROW;
    float* bswz1 = ws + WS_BSWZ1;
    float* bswz2 = ws + WS_BSWZ2;

    // Zero the accumulators + zero-row in one shot (atomics need clean state).
    hipMemsetAsync(ws, 0, (size_t)WS_ZERO_FLOATS * sizeof(float), stream);

    sgh_prep_b1<<<(KOFF * 32 * 32 + 255) / 256, 256, 0, stream>>>(kernel1, bswz1);
    sgh_prep_b2<<<(KOFF * 4 * 32 + 255) / 256, 256, 0, stream>>>(kernel2, bswz2);

    const int conv_blocks = (KOFF * TILES + WPB - 1) / WPB;        // 84375
    sgh_conv1_wmma<<<conv_blocks, 256, 0, stream>>>(feat, bswz1, zrow,
                                                    in_idx, out_idx, mid);

    const int gelu_blocks = (int)(((size_t)NVOX * MSTR / 4 + 255) / 256);
    sgh_gelu<<<gelu_blocks, 256, 0, stream>>>(mid);

    sgh_conv2_wmma<<<conv_blocks, 256, 0, stream>>>(mid, bswz2,
                                                    in_idx, out_idx, out16);

    sgh_repack<<<(NVOX * C_MID + 255) / 256, 256, 0, stream>>>(out16, out);
}